// LegionellaModel_25898652795108
// MI455X (gfx1250) — compile-verified
//
#include <hip/hip_runtime.h>
#include <math.h>

#define NHEAT 12
#define NPIPE 26
#define NBR   3
#define NNODE (NHEAT + NPIPE + NBR)   // 41

// ---- constants: double-precision host math, rounded once to f32 (matches numpy->jax) ----
constexpr double d_pi = 3.14159265358979323846;
constexpr float F_ADV        = 3.04f;                                             // PIPE_V / PIPE_L
constexpr float F_PIPE_OUT   = (float)(d_pi * (0.019 / 2) * (0.019 / 2) * 3.04 * 1000.0);
constexpr float F_SHOWER_OUT = (float)(d_pi * (0.012 / 2) * (0.012 / 2) * 3.04 * 1000.0);
constexpr float F_HPOW       = (float)(5500.0 * 1.89 / (3600.0 * 57.86));
constexpr float F_HDEC       = (float)(5500.0 * 0.1 * 1.89 / (3600.0 * 57.86));
constexpr float F_PIPE_RATE  = (float)(-4.0 * 10.0 / (4186.8 * 0.019 * 997.0 * 3.04));
constexpr float F_BR_RATE    = (float)(-4.0 * 10.0 / (4186.8 * 0.006 * 997.0 * 3.04));
constexpr float F_INV_HNV    = (float)(12.0 / 295.0);                             // 1 / (295/12)
constexpr float F_MIX        = 0.3f;
constexpr float F_ARR_A      = (float)(1800000.0 / 60.0);                         // 30000
constexpr float F_ARR_B      = 6050.0f;
constexpr float F_C2K        = 273.0f;
constexpr float F_SETM1      = 47.0f;                                             // SET_POINT - 1
constexpr float F_SL_CHL_E   = (float)(-0.46 / 60.0);
constexpr float F_SL_CHL_L   = (float)(-0.1 / 60.0);
constexpr float F_SL_RATE_E  = (float)(1.3 / 60.0);
constexpr float F_SL_RATE_L  = (float)(0.06 / 60.0);

__device__ __forceinline__ float ntload(const float* p) {
    return __builtin_nontemporal_load(p);
}
__device__ __forceinline__ void ntstore(float* p, float v) {
    __builtin_nontemporal_store(v, p);
}

// piecewise(temps, UB_T, RATE_T): idx = sum(x < ub) - 1 ; exact strict-< semantics
__device__ __forceinline__ float rateT(float T) {
    return T < 30.f ? 2.3e-05f
         : T < 37.f ? 3.08e-05f
         : T < 42.f ? 4.73e-05f
         : T < 45.f ? 4.82e-05f
         : T < 50.f ? 0.0f
         : T < 55.f ? -5.69e-05f
         : T < 60.f ? -0.00666f
         : T < 65.f ? -0.0333f
         : T < 70.f ? -0.0833f
         :            -0.0741f;
}
__device__ __forceinline__ float rateC(float C) {
    return C < 0.01f ? 0.0f
         : C < 0.15f ? -0.00182f
         : C < 0.35f ? -0.00192f
         :             -0.0231f;
}

__global__ __launch_bounds__(256)
void legionella_fwd(const float* __restrict__ t_ptr,
                    const float* __restrict__ dens,
                    const float* __restrict__ env_p,
                    const float* __restrict__ mlt_p,
                    const float* __restrict__ mlc_p,
                    const float* __restrict__ il_p,
                    const float* __restrict__ sr_p,
                    const float* __restrict__ toc_p,
                    const float* __restrict__ sd_p,
                    float* __restrict__ out,
                    int Bn)
{
    const int b = blockIdx.x * blockDim.x + threadIdx.x;
    if (b >= Bn) return;
    const size_t sb = (size_t)Bn;

    // ---- per-batch scalars (coalesced, one dword each) ----
    const float t   = t_ptr[0];            // uniform -> scalar load
    const float env = ntload(env_p + b);
    const float mlt = ntload(mlt_p + b);
    const float mlc = ntload(mlc_p + b);
    const float il  = ntload(il_p  + b);
    const float sr  = ntload(sr_p  + b);
    const float toc = ntload(toc_p + b);
    const float sd  = ntload(sd_p  + b);

    const float shower_on   = ((t < 1.0f + sd) && (1.0f < t)) ? 1.0f : 0.0f;
    const float branch_rate = F_SHOWER_OUT * shower_on;                 // SHOWER_OUT*on
    const float recirc      = F_PIPE_OUT - F_SHOWER_OUT * shower_on;
    const bool  early       = (t - 1.0f) < 300.0f;
    const float sluff_chl   = (early ? F_SL_CHL_E  : F_SL_CHL_L)  * shower_on;
    const float sluff_rate  = (early ? F_SL_RATE_E : F_SL_RATE_L) * shower_on;
    const float sr_sluff    = sr * sluff_rate;
    const float arrA_toc    = -F_ARR_A * toc;

    const float* pin  = dens + b;
    float*       pout = out  + b;

    // ---- heater block: all 12 nodes x {T,C,L,S} (needed for thermostat + mixing) ----
    float hT[NHEAT], hC[NHEAT], hL[NHEAT], hS[NHEAT];
#pragma unroll
    for (int i = 0; i < NHEAT; ++i) {
        const size_t base = (size_t)(i * 5) * sb;
        hT[i] = ntload(pin + base);
        hC[i] = ntload(pin + base + sb);
        hL[i] = ntload(pin + base + 2 * sb);
        hS[i] = ntload(pin + base + 3 * sb);
    }
    const float top_temp =
        (hT[5] + hT[6] + hT[7] + hT[8] + hT[9] + hT[10] + hT[11]) / 7.0f;
    const bool heater_on = top_temp < F_SETM1;

    // ---- pipe nodes (advection from heater[11]); capture pipe[12] and pipe[25] ----
    float pvT = hT[11], pvC = hC[11], pvL = hL[11], pvS = hS[11];
    float p12T = 0.f, p12C = 0.f, p12L = 0.f, p12S = 0.f;
    float p25T = 0.f, p25C = 0.f, p25L = 0.f, p25S = 0.f;

    for (int j = 0; j < NPIPE; ++j) {
        const int g = NHEAT + j;
        const size_t base = (size_t)(g * 5) * sb;

        // gfx1250 global_prefetch_b8: prime next node's 5 streams
        if (j + 1 < NPIPE) {
            const size_t nb = (size_t)((g + 1) * 5) * sb;
            __builtin_prefetch(pin + nb,          0, 1);
            __builtin_prefetch(pin + nb + sb,     0, 1);
            __builtin_prefetch(pin + nb + 2 * sb, 0, 1);
            __builtin_prefetch(pin + nb + 3 * sb, 0, 1);
            __builtin_prefetch(pin + nb + 4 * sb, 0, 1);
        }

        const float cT = ntload(pin + base);
        const float cC = ntload(pin + base + sb);
        const float cL = ntload(pin + base + 2 * sb);
        const float cS = ntload(pin + base + 3 * sb);
        const float cB = ntload(pin + base + 4 * sb);

        float dT = (pvT - cT) * F_ADV;
        float dC = (pvC - cC) * F_ADV;
        float dL = (pvL - cL) * F_ADV;
        float dS = (pvS - cS) * F_ADV;

        dT += (cT - env) * F_PIPE_RATE;
        const float cd = arrA_toc * expf(-F_ARR_B / (cT + F_C2K));
        dC += cd * cC;
        dC += sluff_chl * cC;
        dL += (rateT(cT) + rateC(cC)) * cL;
        const float srb = sluff_rate * cB;
        dL += sr_sluff * cB;
        dS += srb;
        const float dB = -srb;

        ntstore(pout + base,          dT);
        ntstore(pout + base + sb,     dC);
        ntstore(pout + base + 2 * sb, dL);
        ntstore(pout + base + 3 * sb, dS);
        ntstore(pout + base + 4 * sb, dB);

        if (j == 12)         { p12T = cT; p12C = cC; p12L = cL; p12S = cS; }
        if (j == NPIPE - 1)  { p25T = cT; p25C = cC; p25L = cL; p25S = cS; }
        pvT = cT; pvC = cC; pvL = cL; pvS = cS;
    }

    // ---- branch nodes (advection from pipe[12] at rate branch_rate) ----
    pvT = p12T; pvC = p12C; pvL = p12L; pvS = p12S;
#pragma unroll
    for (int k = 0; k < NBR; ++k) {
        const int g = NHEAT + NPIPE + k;
        const size_t base = (size_t)(g * 5) * sb;

        const float cT = ntload(pin + base);
        const float cC = ntload(pin + base + sb);
        const float cL = ntload(pin + base + 2 * sb);
        const float cS = ntload(pin + base + 3 * sb);
        const float cB = ntload(pin + base + 4 * sb);

        float dT = (pvT - cT) * branch_rate;
        float dC = (pvC - cC) * branch_rate;
        float dL = (pvL - cL) * branch_rate;
        float dS = (pvS - cS) * branch_rate;

        dT += (cT - env) * F_BR_RATE;
        const float cd = arrA_toc * expf(-F_ARR_B / (cT + F_C2K));
        dC += cd * cC;
        dC += sluff_chl * cC;
        dL += (rateT(cT) + rateC(cC)) * cL;
        const float srb = sluff_rate * cB;
        dL += sr_sluff * cB;
        dS += srb;
        const float dB = -srb;

        ntstore(pout + base,          dT);
        ntstore(pout + base + sb,     dC);
        ntstore(pout + base + 2 * sb, dL);
        ntstore(pout + base + 3 * sb, dS);
        ntstore(pout + base + 4 * sb, dB);

        pvT = cT; pvC = cC; pvL = cL; pvS = cS;
    }

    // ---- heater derivatives (mixing + recirc/intake at node 0 + heat flux) ----
#pragma unroll
    for (int i = 0; i < NHEAT; ++i) {
        float mT = 0.f, mC = 0.f, mL = 0.f, mS = 0.f;
        if (i > 0) {
            mT += F_MIX * ((hT[i - 1] - hT[i]) * F_INV_HNV);
            mC += F_MIX * ((hC[i - 1] - hC[i]) * F_INV_HNV);
            mL += F_MIX * ((hL[i - 1] - hL[i]) * F_INV_HNV);
            mS += F_MIX * ((hS[i - 1] - hS[i]) * F_INV_HNV);
        }
        if (i < NHEAT - 1) {
            mT -= F_MIX * ((hT[i] - hT[i + 1]) * F_INV_HNV);
            mC -= F_MIX * ((hC[i] - hC[i + 1]) * F_INV_HNV);
            mL -= F_MIX * ((hL[i] - hL[i + 1]) * F_INV_HNV);
            mS -= F_MIX * ((hS[i] - hS[i + 1]) * F_INV_HNV);
        }
        if (i == 0) {
            mT += recirc * (p25T - hT[0]) * F_INV_HNV + branch_rate * (mlt - hT[0]) * F_INV_HNV;
            mC += recirc * (p25C - hC[0]) * F_INV_HNV + branch_rate * (mlc - hC[0]) * F_INV_HNV;
            mL += recirc * (p25L - hL[0]) * F_INV_HNV + branch_rate * (il  - hL[0]) * F_INV_HNV;
            mS += recirc * (p25S - hS[0]) * F_INV_HNV + branch_rate * (0.f - hS[0]) * F_INV_HNV;
        }
        const float flux = heater_on ? ((i == 3 || i == 9) ? F_HPOW : 0.0f) : -F_HDEC;
        mT += flux;

        const float cd = arrA_toc * expf(-F_ARR_B / (hT[i] + F_C2K));
        mC += cd * hC[i];
        mL += (rateT(hT[i]) + rateC(hC[i])) * hL[i];
        // IS_PIPE == 0 for heater nodes: no sluff terms; bf_der == 0

        const size_t base = (size_t)(i * 5) * sb;
        ntstore(pout + base,          mT);
        ntstore(pout + base + sb,     mC);
        ntstore(pout + base + 2 * sb, mL);
        ntstore(pout + base + 3 * sb, mS);
        ntstore(pout + base + 4 * sb, 0.0f);
    }
}

extern "C" void kernel_launch(void* const* d_in, const int* in_sizes, int n_in,
                              void* d_out, int out_size, void* d_ws, size_t ws_size,
                              hipStream_t stream) {
    const float* t    = (const float*)d_in[0];
    const float* dens = (const float*)d_in[1];
    const float* env  = (const float*)d_in[2];
    const float* mlt  = (const float*)d_in[3];
    const float* mlc  = (const float*)d_in[4];
    const float* il   = (const float*)d_in[5];
    const float* sr   = (const float*)d_in[6];
    const float* toc  = (const float*)d_in[7];
    const float* sd   = (const float*)d_in[8];
    float* out = (float*)d_out;

    const int Bn = in_sizes[2];             // env_temp has shape (B,)
    const int block = 256;                  // 8 wave32 waves per workgroup
    const int grid  = (Bn + block - 1) / block;
    legionella_fwd<<<grid, block, 0, stream>>>(t, dens, env, mlt, mlc, il, sr, toc, sd, out, Bn);
}